// QuantumLayer_10316511445518
// MI455X (gfx1250) — compile-verified
//
#include <hip/hip_runtime.h>

typedef __attribute__((ext_vector_type(2))) float v2f;
typedef __attribute__((ext_vector_type(8))) float v8f;

#define NQ 16

// One wave32 per batch row. qm = lane & 15 is the qubit index.
// Closed form (MPS bond-dim-2 contraction of the reference circuit):
//   <Z_k> = cos(theta) * prod_{j<=k} cos(x_j)
//         - sin(theta)cos(phi) * sin(x_k) * (k<15 ? sin(x_{k+1}) : 1)
// Realized as diag( A[16x4] x B[4x16] ) via V_WMMA_F32_16X16X4_F32:
//   A: lane m (<16): {d0_m, 0};  lane m+16: {0, d1_m}   (d0/d1 = (1±Dprev)/2)
//   B: lane n (<16): {q00_n, 0}; lane n+16: {0, q11_n}
//   D[m,n] = d0_m*q00_n + d1_m*q11_n ; diagonal = <Z_k>.
__global__ __launch_bounds__(256) void qlayer_wmma_kernel(const float* __restrict__ x,
                                                          float* __restrict__ out,
                                                          int batch) {
  const int lane = threadIdx.x & 31;
  const int wave = threadIdx.x >> 5;
  const int b = blockIdx.x * 8 + wave;
  if (b >= batch) return;  // wave-uniform; grid exactly covers batch anyway

  const int qm = lane & 15;

  // full-angle cos/sin of x[b, qm]
  const float xv = x[b * NQ + qm];
  const float c = cosf(xv);
  const float s = sinf(xv);

  // W_k = sin(x_{k+1}) for k<15 else 1  (fetch from neighbor lane)
  const float s_next = __shfl(s, (lane & 16) | ((qm + 1) & 15), 32);
  const float W = (qm == 15) ? 1.0f : s_next;

  // inclusive prefix product of cos(x_j) over j<=qm (log-step shuffle scan;
  // both 16-lane halves hold identical data, so cross-half reads are benign)
  float p = c;
#pragma unroll
  for (int off = 1; off < 16; off <<= 1) {
    float t = __shfl(p, (lane & 16) | ((qm - off) & 15), 32);
    if (qm >= off) p *= t;
  }
  // exclusive prefix D_{k-1} = prod_{j<k} cos(x_j)
  const float t0 = __shfl(p, (lane & 16) | ((qm - 1) & 15), 32);
  const float Dprev = (qm == 0) ? 1.0f : t0;

  const float CT = 0.98006657784124163f;  // cos(theta = 0.2)
  const float K2 = 0.19767681165408385f;  // sin(0.2) * cos(phi = 0.1)

  const float d0 = 0.5f * (1.0f + Dprev);
  const float d1 = 0.5f * (1.0f - Dprev);
  const float q00 =  CT * c - K2 * W * s;
  const float q11 = -CT * c - K2 * W * s;

  const bool hi = lane >= 16;
  v2f A, B;
  A.x = hi ? 0.0f : d0;   A.y = hi ? d1  : 0.0f;
  B.x = hi ? 0.0f : q00;  B.y = hi ? q11 : 0.0f;

  v8f C = {};
  // D[16,16] = A[16,4] x B[4,16] + 0  -> v_wmma_f32_16x16x4_f32
  v8f D = __builtin_amdgcn_wmma_f32_16x16x4_f32(
      /*neg_a=*/false, A, /*neg_b=*/false, B,
      /*c_mod=*/(short)0, C, /*reuse_a=*/false, /*reuse_b=*/false);

  // Branch-free diagonal extraction per C/D layout:
  //   lanes 0..7   hold D[e, lane]      -> diagonal when e == lane      (k = lane)
  //   lanes 24..31 hold D[e+8, lane-16] -> diagonal when e == lane - 24 (k = lane-16)
  const int e = lane & 7;
  float val = 0.0f;
#pragma unroll
  for (int i = 0; i < 8; ++i) {
    val += (e == i) ? D[i] : 0.0f;  // flat, independent cndmask selects
  }
  const bool writer = (lane < 8) || (lane >= 24);
  const int k = (lane < 8) ? lane : (lane - 16);
  if (writer) out[b * NQ + k] = val;
}

extern "C" void kernel_launch(void* const* d_in, const int* in_sizes, int n_in,
                              void* d_out, int out_size, void* d_ws, size_t ws_size,
                              hipStream_t stream) {
  const float* x = (const float*)d_in[0];
  float* out = (float*)d_out;
  const int batch = in_sizes[0] / NQ;  // 512
  const int wavesPerBlock = 8;         // 256 threads = 8 wave32
  dim3 block(32 * wavesPerBlock);
  dim3 grid((batch + wavesPerBlock - 1) / wavesPerBlock);
  qlayer_wmma_kernel<<<grid, block, 0, stream>>>(x, out, batch);
}